// DelaunayGNN_5600637354242
// MI455X (gfx1250) — compile-verified
//
#include <hip/hip_runtime.h>

#define N_NODES 50000
#define N_EDGES 800000
#define E2      (N_EDGES + N_NODES)
#define N_GRAPHS 32
#define HID 128
#define OUT_DIM 40
#define EPS 1e-5f
#define SLOPE 0.2f

typedef __attribute__((ext_vector_type(2))) float v2f;
typedef __attribute__((ext_vector_type(8))) float v8f;

// ---------- helpers ----------
__device__ __forceinline__ unsigned f2ord(float f) {
  unsigned u = __float_as_uint(f);
  return (u & 0x80000000u) ? ~u : (u | 0x80000000u);
}
__device__ __forceinline__ float ord2f(unsigned u) {
  return __uint_as_float((u & 0x80000000u) ? (u ^ 0x80000000u) : ~u);
}

__global__ void fill_u32_kernel(unsigned* p, unsigned v, long n) {
  long i = (long)blockIdx.x * blockDim.x + threadIdx.x;
  long st = (long)gridDim.x * blockDim.x;
  for (; i < n; i += st) p[i] = v;
}

// ---------- self-loop attr (mean of incoming edge_attr per node) ----------
__global__ void loop_accum_kernel(const int* __restrict__ eidx,
                                  const float* __restrict__ eattr,
                                  float* __restrict__ cnt, float* __restrict__ tot) {
  int e = blockIdx.x * blockDim.x + threadIdx.x;
  if (e >= N_EDGES) return;
  int d = eidx[N_EDGES + e];
  atomicAdd(cnt + d, 1.0f);
  atomicAdd(tot + d * 3 + 0, eattr[e * 3 + 0]);
  atomicAdd(tot + d * 3 + 1, eattr[e * 3 + 1]);
  atomicAdd(tot + d * 3 + 2, eattr[e * 3 + 2]);
}
__global__ void loop_final_kernel(float* __restrict__ tot, const float* __restrict__ cnt) {
  int i = blockIdx.x * blockDim.x + threadIdx.x;
  if (i >= N_NODES * 3) return;
  float c = cnt[i / 3];
  tot[i] = tot[i] / fmaxf(c, 1.0f);
}

// ---------- layer0 transform: x[N,3] -> xl,xr [N,128] ----------
__global__ void xform0_kernel(const float* __restrict__ x,
                              const float* __restrict__ W0l, const float* __restrict__ W0r,
                              float* __restrict__ xl, float* __restrict__ xr) {
  int idx = blockIdx.x * blockDim.x + threadIdx.x;
  if (idx >= N_NODES * HID) return;
  int n = idx >> 7, c = idx & 127;
  float x0 = x[n * 3], x1 = x[n * 3 + 1], x2 = x[n * 3 + 2];
  xl[idx] = x0 * W0l[c] + x1 * W0l[HID + c] + x2 * W0l[2 * HID + c];
  xr[idx] = x0 * W0r[c] + x1 * W0r[HID + c] + x2 * W0r[2 * HID + c];
}

// ---------- fp32 WMMA GEMM: C[M,128] = A[M,128] @ B[128,128] ----------
// One wave per 16x16 output tile; 8 waves/block cover all 128 columns.
__global__ __launch_bounds__(256) void gemm_wmma_f32(const float* __restrict__ A,
                                                     const float* __restrict__ B,
                                                     float* __restrict__ C, int M) {
  int wave = threadIdx.x >> 5;
  int lane = threadIdx.x & 31;
  int m0 = blockIdx.x * 16;
  int n0 = wave * 16;
  int r  = lane & 15;   // row (A) / col (B,C) within tile
  int kh = lane >> 4;   // K-half select
  int arow = m0 + r; if (arow >= M) arow = M - 1;
  const float* Ap = A + (long)arow * HID;
  v8f acc = {};
#pragma unroll 4
  for (int k = 0; k < HID; k += 4) {
    v2f a, b;
    a.x = Ap[k + 2 * kh];
    a.y = Ap[k + 2 * kh + 1];
    b.x = B[(k + 2 * kh) * HID + n0 + r];
    b.y = B[(k + 2 * kh + 1) * HID + n0 + r];
    acc = __builtin_amdgcn_wmma_f32_16x16x4_f32(false, a, false, b, (short)0, acc,
                                                false, false);
  }
#pragma unroll
  for (int v = 0; v < 8; ++v) {
    int row = m0 + v + 8 * kh;
    if (row < M) C[(long)row * HID + n0 + r] = acc[v];
  }
}

// ---------- edge pass A: logits + segment-max (one wave per edge) ----------
__global__ __launch_bounds__(256) void edge_pass_a(const float* __restrict__ xl,
                                                   const float* __restrict__ xr,
                                                   const int* __restrict__ eidx,
                                                   const float* __restrict__ eattr,
                                                   const float* __restrict__ lattr,
                                                   const float* __restrict__ We,
                                                   const float* __restrict__ av,
                                                   float* __restrict__ ebuf,
                                                   unsigned* __restrict__ lmaxb) {
  __shared__ float sWe[3 * HID];
  __shared__ float sA[HID];
  for (int i = threadIdx.x; i < 3 * HID; i += blockDim.x) sWe[i] = We[i];
  for (int i = threadIdx.x; i < HID; i += blockDim.x) sA[i] = av[i];
  __syncthreads();
  int lane = threadIdx.x & 31;
  int wid = blockIdx.x * (blockDim.x >> 5) + (threadIdx.x >> 5);
  int nw = gridDim.x * (blockDim.x >> 5);
  for (int e = wid; e < E2; e += nw) {
    int s, d; float e0, e1, e2;
    if (e < N_EDGES) {
      s = eidx[e]; d = eidx[N_EDGES + e];
      e0 = eattr[e * 3]; e1 = eattr[e * 3 + 1]; e2 = eattr[e * 3 + 2];
    } else {
      s = e - N_EDGES; d = s;
      e0 = lattr[s * 3]; e1 = lattr[s * 3 + 1]; e2 = lattr[s * 3 + 2];
    }
    int c0 = lane * 4;
    const float4 xs = *(const float4*)(xl + (long)s * HID + c0);
    const float4 xd = *(const float4*)(xr + (long)d * HID + c0);
    const float* xsp = (const float*)&xs;
    const float* xdp = (const float*)&xd;
    float part = 0.0f;
#pragma unroll
    for (int j = 0; j < 4; ++j) {
      int c = c0 + j;
      float ec = e0 * sWe[c] + e1 * sWe[HID + c] + e2 * sWe[2 * HID + c];
      float m = xsp[j] + xdp[j] + ec;
      m = m > 0.0f ? m : SLOPE * m;
      part += m * sA[c];
    }
#pragma unroll
    for (int off = 16; off > 0; off >>= 1) part += __shfl_xor(part, off, 32);
    if (lane == 0) {
      ebuf[e] = part;
      atomicMax(lmaxb + d, f2ord(part));
    }
  }
}

// ---------- edge pass B: exp + segment-sum denominator ----------
__global__ void edge_pass_b(float* __restrict__ ebuf, const int* __restrict__ eidx,
                            const unsigned* __restrict__ lmaxb, float* __restrict__ den) {
  int e = blockIdx.x * blockDim.x + threadIdx.x;
  if (e >= E2) return;
  int d = (e < N_EDGES) ? eidx[N_EDGES + e] : (e - N_EDGES);
  float ex = expf(ebuf[e] - ord2f(lmaxb[d]));
  ebuf[e] = ex;
  atomicAdd(den + d, ex);
}

// ---------- edge pass C: weighted scatter (one wave per edge) ----------
__global__ __launch_bounds__(256) void edge_pass_c(const float* __restrict__ xl,
                                                   const int* __restrict__ eidx,
                                                   const float* __restrict__ ebuf,
                                                   const float* __restrict__ den,
                                                   float* __restrict__ acc) {
  int lane = threadIdx.x & 31;
  int wid = blockIdx.x * (blockDim.x >> 5) + (threadIdx.x >> 5);
  int nw = gridDim.x * (blockDim.x >> 5);
  for (int e = wid; e < E2; e += nw) {
    int s, d;
    if (e < N_EDGES) { s = eidx[e]; d = eidx[N_EDGES + e]; }
    else             { s = e - N_EDGES; d = s; }
    float alpha = ebuf[e] / den[d];
    int c0 = lane * 4;
    const float4 v = *(const float4*)(xl + (long)s * HID + c0);
    float* ap = acc + (long)d * HID + c0;
    atomicAdd(ap + 0, alpha * v.x);
    atomicAdd(ap + 1, alpha * v.y);
    atomicAdd(ap + 2, alpha * v.z);
    atomicAdd(ap + 3, alpha * v.w);
  }
}

// ---------- relu(acc+bias) -> h, accumulate per-channel sum / sumsq ----------
__global__ __launch_bounds__(256) void relu_stats_kernel(const float* __restrict__ acc,
                                                         const float* __restrict__ bias,
                                                         float* __restrict__ h,
                                                         float* __restrict__ sums,
                                                         float* __restrict__ sumsq) {
  __shared__ float ls[HID], lq[HID];
  for (int i = threadIdx.x; i < HID; i += blockDim.x) { ls[i] = 0.f; lq[i] = 0.f; }
  __syncthreads();
  int ch = threadIdx.x & 127;
  int npb = blockDim.x >> 7;                    // nodes per block per iter
  int n0 = blockIdx.x * npb + (threadIdx.x >> 7);
  float s = 0.f, q = 0.f;
  for (int n = n0; n < N_NODES; n += gridDim.x * npb) {
    float v = acc[(long)n * HID + ch] + bias[ch];
    v = v > 0.f ? v : 0.f;
    h[(long)n * HID + ch] = v;
    s += v; q += v * v;
  }
  atomicAdd(ls + ch, s);
  atomicAdd(lq + ch, q);
  __syncthreads();
  if (threadIdx.x < HID) {
    atomicAdd(sums + threadIdx.x, ls[threadIdx.x]);
    atomicAdd(sumsq + threadIdx.x, lq[threadIdx.x]);
  }
}

__global__ void bn_norm_kernel(float* __restrict__ h, const float* __restrict__ sums,
                               const float* __restrict__ sumsq,
                               const float* __restrict__ g, const float* __restrict__ b) {
  int idx = blockIdx.x * blockDim.x + threadIdx.x;
  if (idx >= N_NODES * HID) return;
  int ch = idx & 127;
  float mu = sums[ch] * (1.0f / N_NODES);
  float var = sumsq[ch] * (1.0f / N_NODES) - mu * mu;
  h[idx] = g[ch] * (h[idx] - mu) * rsqrtf(var + EPS) + b[ch];
}

// ---------- segment-max pooling over batch ----------
__global__ void pool_kernel(const float* __restrict__ h, const int* __restrict__ batch,
                            unsigned* __restrict__ poolb) {
  int idx = blockIdx.x * blockDim.x + threadIdx.x;
  if (idx >= N_NODES * HID) return;
  int n = idx >> 7, ch = idx & 127;
  atomicMax(poolb + batch[n] * HID + ch, f2ord(h[idx]));
}
__global__ void pool_decode_kernel(const unsigned* __restrict__ poolb,
                                   float* __restrict__ poolf) {
  int i = blockIdx.x * blockDim.x + threadIdx.x;
  if (i < N_GRAPHS * HID) poolf[i] = ord2f(poolb[i]);
}

// ---------- FC head: out = relu(z + fc1_b) @ fc2_W + fc2_b ----------
__global__ __launch_bounds__(128) void head_kernel(const float* __restrict__ z,
                                                   const float* __restrict__ fc1b,
                                                   const float* __restrict__ fc2W,
                                                   const float* __restrict__ fc2b,
                                                   float* __restrict__ out) {
  __shared__ float rz[HID];
  int g = blockIdx.x, t = threadIdx.x;
  if (t < HID) {
    float v = z[g * HID + t] + fc1b[t];
    rz[t] = v > 0.f ? v : 0.f;
  }
  __syncthreads();
  if (t < OUT_DIM) {
    float s = fc2b[t];
#pragma unroll 8
    for (int k = 0; k < HID; ++k) s += rz[k] * fc2W[k * OUT_DIM + t];
    out[g * OUT_DIM + t] = s;
  }
}

// ================= host-side launch =================
static inline void fill0(void* p, long n_u32, hipStream_t s) {
  int grid = (int)((n_u32 + 255) / 256); if (grid > 4096) grid = 4096;
  fill_u32_kernel<<<grid, 256, 0, s>>>((unsigned*)p, 0u, n_u32);
}

extern "C" void kernel_launch(void* const* d_in, const int* in_sizes, int n_in,
                              void* d_out, int out_size, void* d_ws, size_t ws_size,
                              hipStream_t stream) {
  const float* x     = (const float*)d_in[0];
  const float* eattr = (const float*)d_in[1];
  const int*   eidx  = (const int*)d_in[2];
  const int*   batch = (const int*)d_in[3];
  const float* W0l   = (const float*)d_in[4];
  const float* W0r   = (const float*)d_in[5];
  const float* We0   = (const float*)d_in[6];
  const float* a0    = (const float*)d_in[7];
  const float* b0    = (const float*)d_in[8];
  const float* bn0g  = (const float*)d_in[9];
  const float* bn0b  = (const float*)d_in[10];
  const float* Wl    = (const float*)d_in[11];  // [2,128,128]
  const float* Wel   = (const float*)d_in[12];  // [2,3,128]
  const float* al    = (const float*)d_in[13];  // [2,128]
  const float* bl    = (const float*)d_in[14];
  const float* bng   = (const float*)d_in[15];
  const float* bnb   = (const float*)d_in[16];
  const float* fc1W  = (const float*)d_in[17];
  const float* fc1b  = (const float*)d_in[18];
  const float* fc2W  = (const float*)d_in[19];
  const float* fc2b  = (const float*)d_in[20];
  float* out = (float*)d_out;

  char* ws = (char*)d_ws;
  auto carve = [&](size_t bytes) -> char* {
    char* p = ws;
    ws += (bytes + 255) & ~(size_t)255;
    return p;
  };
  float*    h     = (float*)carve((size_t)N_NODES * HID * 4);
  float*    xl    = (float*)carve((size_t)N_NODES * HID * 4);
  float*    xr    = (float*)carve((size_t)N_NODES * HID * 4);
  float*    acc   = (float*)carve((size_t)N_NODES * HID * 4);
  unsigned* lmaxb = (unsigned*)carve((size_t)N_NODES * 4);
  float*    den   = (float*)carve((size_t)N_NODES * 4);
  float*    ebuf  = (float*)carve((size_t)E2 * 4);
  float*    cnt   = (float*)carve((size_t)N_NODES * 4);
  float*    lattr = (float*)carve((size_t)N_NODES * 3 * 4);
  float*    sums  = (float*)carve(HID * 4);
  float*    sumsq = (float*)carve(HID * 4);
  unsigned* poolb = (unsigned*)carve((size_t)N_GRAPHS * HID * 4);
  float*    poolf = (float*)carve((size_t)N_GRAPHS * HID * 4);
  float*    z     = (float*)carve((size_t)N_GRAPHS * HID * 4);

  // self-loop attributes
  fill0(cnt, N_NODES, stream);
  fill0(lattr, (long)N_NODES * 3, stream);
  loop_accum_kernel<<<(N_EDGES + 255) / 256, 256, 0, stream>>>(eidx, eattr, cnt, lattr);
  loop_final_kernel<<<(N_NODES * 3 + 255) / 256, 256, 0, stream>>>(lattr, cnt);

  // layer 0 input transforms
  xform0_kernel<<<(N_NODES * HID + 255) / 256, 256, 0, stream>>>(x, W0l, W0r, xl, xr);

  for (int L = 0; L < 3; ++L) {
    const float* Wecur = (L == 0) ? We0 : (Wel + (size_t)(L - 1) * 3 * HID);
    const float* acur  = (L == 0) ? a0  : (al  + (size_t)(L - 1) * HID);
    const float* bcur  = (L == 0) ? b0  : (bl  + (size_t)(L - 1) * HID);
    const float* gcur  = (L == 0) ? bn0g : (bng + (size_t)(L - 1) * HID);
    const float* btcur = (L == 0) ? bn0b : (bnb + (size_t)(L - 1) * HID);
    const float *XL, *XR;
    if (L == 0) { XL = xl; XR = xr; }
    else {
      gemm_wmma_f32<<<(N_NODES + 15) / 16, 256, 0, stream>>>(
          h, Wl + (size_t)(L - 1) * HID * HID, xl, N_NODES);
      XL = xl; XR = xl;   // share_weights=True
    }
    fill0(lmaxb, N_NODES, stream);
    fill0(den, N_NODES, stream);
    fill0(acc, (long)N_NODES * HID, stream);
    fill0(sums, HID, stream);
    fill0(sumsq, HID, stream);
    edge_pass_a<<<4096, 256, 0, stream>>>(XL, XR, eidx, eattr, lattr, Wecur, acur,
                                          ebuf, lmaxb);
    edge_pass_b<<<(E2 + 255) / 256, 256, 0, stream>>>(ebuf, eidx, lmaxb, den);
    edge_pass_c<<<4096, 256, 0, stream>>>(XL, eidx, ebuf, den, acc);
    relu_stats_kernel<<<512, 256, 0, stream>>>(acc, bcur, h, sums, sumsq);
    bn_norm_kernel<<<(N_NODES * HID + 255) / 256, 256, 0, stream>>>(h, sums, sumsq,
                                                                    gcur, btcur);
  }

  // pooling + FC head
  fill0(poolb, (long)N_GRAPHS * HID, stream);
  pool_kernel<<<(N_NODES * HID + 255) / 256, 256, 0, stream>>>(h, batch, poolb);
  pool_decode_kernel<<<(N_GRAPHS * HID + 255) / 256, 256, 0, stream>>>(poolb, poolf);
  gemm_wmma_f32<<<(N_GRAPHS + 15) / 16, 256, 0, stream>>>(poolf, fc1W, z, N_GRAPHS);
  head_kernel<<<N_GRAPHS, 128, 0, stream>>>(z, fc1b, fc2W, fc2b, out);
}